// ParallelDroplessMLP_56392920596548
// MI455X (gfx1250) — compile-verified
//
#include <hip/hip_runtime.h>
#include <math.h>

// ---------------------------------------------------------------------------
// Dropless MoE MLP (SwiGLU) for MI455X / gfx1250, fp32 WMMA path.
//   out[t] = sum_k ew[t,k] * ( silu(x[t] @ w1[e]) * (x[t] @ w3[e]) ) @ w2[e]
//   plus batch_size_per_expert histogram (int32, appended to d_out).
// ---------------------------------------------------------------------------

#define D_MODEL   1024
#define HIDDEN    1024
#define N_EXPERTS 8
#define TOP_K     2
#define N_TOKENS  4096
#define N_ASSIGN  (N_TOKENS * TOP_K)            // 8192 routed assignments
#define TILE_M    32                            // tokens per GEMM tile
#define MAX_TILES (N_ASSIGN / TILE_M + N_EXPERTS)  // 264 (worst-case padding)
#define LDS_STRIDE 1028                         // pad 4: 16B-aligned rows, bank stride 4

// workspace layout (int32 units)
#define W_NTILES  0
#define W_TILE_E  16
#define W_TILE_RS (16 + MAX_TILES)              // 280
#define W_ROWS    (16 + 2 * MAX_TILES)          // 544; rows[] = MAX_TILES*TILE_M ints

typedef float v2f __attribute__((ext_vector_type(2)));
typedef float v8f __attribute__((ext_vector_type(8)));

// ---------------------------------------------------------------------------
__global__ __launch_bounds__(256) void moe_zero(float4* __restrict__ out4) {
    size_t i = (size_t)blockIdx.x * 256 + threadIdx.x;   // 4096*256 threads * 4 floats
    out4[i] = make_float4(0.f, 0.f, 0.f, 0.f);
}

// ---------------------------------------------------------------------------
// Routing: histogram, tile-aligned offsets, scatter assignment ids per expert.
__global__ __launch_bounds__(256) void moe_route(const int* __restrict__ idx,
                                                 int* __restrict__ ws,
                                                 int* __restrict__ hist_out) {
    __shared__ int cnt[N_EXPERTS];
    __shared__ int aoff[N_EXPERTS + 1];
    __shared__ int cur[N_EXPERTS];
    const int tid = threadIdx.x;
    if (tid < N_EXPERTS) { cnt[tid] = 0; cur[tid] = 0; }
    __syncthreads();

    for (int i = tid; i < N_ASSIGN; i += 256) atomicAdd(&cnt[idx[i]], 1);
    int* rows = ws + W_ROWS;
    for (int i = tid; i < MAX_TILES * TILE_M; i += 256) rows[i] = -1;  // padding
    __syncthreads();

    if (tid == 0) {
        int nt = 0, off = 0;
        for (int e = 0; e < N_EXPERTS; ++e) {
            aoff[e] = off;
            const int tiles = (cnt[e] + TILE_M - 1) / TILE_M;
            for (int t = 0; t < tiles; ++t) {
                ws[W_TILE_E + nt]  = e;
                ws[W_TILE_RS + nt] = off + t * TILE_M;
                ++nt;
            }
            off += tiles * TILE_M;
            hist_out[e] = cnt[e];                 // batch_size_per_expert
        }
        aoff[N_EXPERTS] = off;
        ws[W_NTILES] = nt;
    }
    __syncthreads();

    for (int i = tid; i < N_ASSIGN; i += 256) {
        const int e = idx[i];
        const int p = atomicAdd(&cur[e], 1);
        rows[aoff[e] + p] = i;                    // assignment id; token = i>>1
    }
}

// ---------------------------------------------------------------------------
// Grouped GEMM: one block = one (expert, 32-token) tile, 512 threads = 16 waves.
// Wave w owns hidden/model columns [w*64, w*64+64), split into 4 col-tiles of 16.
__global__ __launch_bounds__(512) void moe_gemm(const float* __restrict__ x,
                                                const float* __restrict__ ew,
                                                const float* __restrict__ w1,
                                                const float* __restrict__ w3,
                                                const float* __restrict__ w2,
                                                const int*   __restrict__ ws,
                                                float* __restrict__ out) {
    __shared__ float smem[TILE_M * LDS_STRIDE];   // x tile, reused for h tile
    __shared__ int   s_tok[TILE_M];
    __shared__ float s_wt[TILE_M];

    const int ntiles = ws[W_NTILES];
    const int bid = blockIdx.x;
    if (bid >= ntiles) return;
    const int e  = ws[W_TILE_E + bid];
    const int rs = ws[W_TILE_RS + bid];
    const int* rows = ws + W_ROWS;

    const int tid = threadIdx.x;
    if (tid < TILE_M) {
        const int aid = rows[rs + tid];
        s_tok[tid] = (aid >= 0) ? (aid >> 1) : -1;
        s_wt[tid]  = (aid >= 0) ? ew[aid] : 0.0f;
    }
    __syncthreads();

    // stage x tile [32 x 1024] into LDS with 128-bit loads/stores
    for (int l = tid; l < TILE_M * (D_MODEL / 4); l += 512) {
        const int r  = l >> 8;                    // 256 float4 per row
        const int c4 = l & 255;
        const int t  = s_tok[r];
        float4 v = make_float4(0.f, 0.f, 0.f, 0.f);
        if (t >= 0) v = ((const float4*)(x + (size_t)t * D_MODEL))[c4];
        *(float4*)&smem[r * LDS_STRIDE + c4 * 4] = v;
    }
    __syncthreads();

    const int wave = tid >> 5;
    const int lane = tid & 31;
    const int half = lane >> 4;          // lanes 16-31 handle K+2/+3, M+8
    const int l16  = lane & 15;
    const int n0   = wave * 64;

    const size_t ebase = (size_t)e * D_MODEL * HIDDEN;
    const float* w1p = w1 + ebase + n0 + l16;
    const float* w3p = w3 + ebase + n0 + l16;
    const float* w2p = w2 + ebase + n0 + l16;

    // -------- GEMM1: h1 = x@w1, h3 = x@w3 (K = D_MODEL) --------
    v8f acc1[2][4], acc3[2][4];
#pragma unroll
    for (int rt = 0; rt < 2; ++rt)
#pragma unroll
        for (int ct = 0; ct < 4; ++ct) { acc1[rt][ct] = (v8f)0.0f; acc3[rt][ct] = (v8f)0.0f; }

#pragma unroll 2
    for (int k0 = 0; k0 < D_MODEL; k0 += 4) {
        const int ka = k0 + 2 * half;
        const v2f a0 = *(const v2f*)&smem[(l16)      * LDS_STRIDE + ka];
        const v2f a1 = *(const v2f*)&smem[(16 + l16) * LDS_STRIDE + ka];
#pragma unroll
        for (int ct = 0; ct < 4; ++ct) {
            const int off = ka * HIDDEN + ct * 16;
            v2f b1, b3;
            b1.x = w1p[off]; b1.y = w1p[off + HIDDEN];
            b3.x = w3p[off]; b3.y = w3p[off + HIDDEN];
            acc1[0][ct] = __builtin_amdgcn_wmma_f32_16x16x4_f32(false, a0, false, b1, (short)0, acc1[0][ct], false, false);
            acc1[1][ct] = __builtin_amdgcn_wmma_f32_16x16x4_f32(false, a1, false, b1, (short)0, acc1[1][ct], false, false);
            acc3[0][ct] = __builtin_amdgcn_wmma_f32_16x16x4_f32(false, a0, false, b3, (short)0, acc3[0][ct], false, false);
            acc3[1][ct] = __builtin_amdgcn_wmma_f32_16x16x4_f32(false, a1, false, b3, (short)0, acc3[1][ct], false, false);
        }
    }
    __syncthreads();   // all waves done reading x tile

    // -------- SwiGLU in registers, write h into (aliased) LDS --------
    // silu(g) = g * sigmoid(g); use hw v_rcp_f32 instead of IEEE divide.
#pragma unroll
    for (int rt = 0; rt < 2; ++rt)
#pragma unroll
        for (int ct = 0; ct < 4; ++ct)
#pragma unroll
            for (int j = 0; j < 8; ++j) {
                const float g   = acc1[rt][ct][j];
                const float sig = __builtin_amdgcn_rcpf(1.0f + __expf(-g));
                const float h   = g * sig * acc3[rt][ct][j];
                const int m = rt * 16 + half * 8 + j;          // C-layout row
                const int c = n0 + ct * 16 + l16;              // C-layout col
                smem[m * LDS_STRIDE + c] = h;
            }
    __syncthreads();

    // -------- GEMM2: y = h@w2 (K = HIDDEN) --------
    v8f acc2[2][4];
#pragma unroll
    for (int rt = 0; rt < 2; ++rt)
#pragma unroll
        for (int ct = 0; ct < 4; ++ct) acc2[rt][ct] = (v8f)0.0f;

#pragma unroll 2
    for (int k0 = 0; k0 < HIDDEN; k0 += 4) {
        const int ka = k0 + 2 * half;
        const v2f a0 = *(const v2f*)&smem[(l16)      * LDS_STRIDE + ka];
        const v2f a1 = *(const v2f*)&smem[(16 + l16) * LDS_STRIDE + ka];
#pragma unroll
        for (int ct = 0; ct < 4; ++ct) {
            const int off = ka * D_MODEL + ct * 16;
            v2f b2;
            b2.x = w2p[off]; b2.y = w2p[off + D_MODEL];
            acc2[0][ct] = __builtin_amdgcn_wmma_f32_16x16x4_f32(false, a0, false, b2, (short)0, acc2[0][ct], false, false);
            acc2[1][ct] = __builtin_amdgcn_wmma_f32_16x16x4_f32(false, a1, false, b2, (short)0, acc2[1][ct], false, false);
        }
    }

    // -------- epilogue: gate-weight scale + atomic accumulate --------
#pragma unroll
    for (int rt = 0; rt < 2; ++rt)
#pragma unroll
        for (int j = 0; j < 8; ++j) {
            const int m = rt * 16 + half * 8 + j;
            const int t = s_tok[m];
            if (t < 0) continue;
            const float wt = s_wt[m];
#pragma unroll
            for (int ct = 0; ct < 4; ++ct) {
                const int c = n0 + ct * 16 + l16;
                atomicAdd(&out[(size_t)t * D_MODEL + c], wt * acc2[rt][ct][j]);
            }
        }
}

// ---------------------------------------------------------------------------
extern "C" void kernel_launch(void* const* d_in, const int* in_sizes, int n_in,
                              void* d_out, int out_size, void* d_ws, size_t ws_size,
                              hipStream_t stream) {
    const float* x   = (const float*)d_in[0];
    const float* ewt = (const float*)d_in[1];
    const int*   idx = (const int*)d_in[2];
    const float* w1  = (const float*)d_in[3];
    const float* w3  = (const float*)d_in[4];
    const float* w2  = (const float*)d_in[5];

    float* out  = (float*)d_out;
    int*   hist = (int*)(out + (size_t)N_TOKENS * D_MODEL);  // int32 tail of d_out
    int*   wsi  = (int*)d_ws;

    moe_zero<<<4096, 256, 0, stream>>>((float4*)out);
    moe_route<<<1, 256, 0, stream>>>(idx, wsi, hist);
    moe_gemm<<<MAX_TILES, 512, 0, stream>>>(x, ewt, w1, w3, w2, wsi, out);
}